// GAT_3350074491117
// MI455X (gfx1250) — compile-verified
//
#include <hip/hip_runtime.h>
#include <math.h>

// ---------------------------------------------------------------------------
// 2-layer GAT for MI455X (gfx1250).
//   Dense projections: fp32 WMMA (v_wmma_f32_16x16x4_f32), LDS-staged A tiles.
//   Edge softmax+aggregate: 3 atomic passes (max / exp-sum / weighted scatter).
// Edge-phase is HBM/atomic bound (~0.9 GB per layer at 23.3 TB/s); GEMMs are
// tiny (2.5 GFLOP), so fp32 WMMA keeps full precision at no cost.
// ---------------------------------------------------------------------------

#define NEG_SLOPE 0.2f

typedef float v2f __attribute__((ext_vector_type(2)));
typedef float v8f __attribute__((ext_vector_type(8)));

// Order-preserving float <-> uint transform so atomicMax(u32) == float max,
// including negatives (values here can be negative post leaky_relu).
__device__ __forceinline__ unsigned f2ord(float f) {
  unsigned u = __float_as_uint(f);
  return (u & 0x80000000u) ? ~u : (u | 0x80000000u);
}
__device__ __forceinline__ float ord2f(unsigned u) {
  return __uint_as_float((u & 0x80000000u) ? (u & 0x7FFFFFFFu) : ~u);
}

// ---------------------------------------------------------------------------
// C[M x NC] = A[M x 128] * B[128 x NC], row-major, fp32, via WMMA 16x16x4.
// Block = 128 threads = 4 waves. One 16-row A tile shared via LDS; wave w
// computes column tile (blockIdx.x*4 + w). LDS row stride 132 dwords ->
// lane l reads banks {4l,4l+1}: conflict-free.
// A-frag layout (ISA 7.12.2, 32-bit A 16x4): lanes 0-15 hold M=0..15,
//   v0 = K = k0 + (lane<16 ? 0 : 2), v1 = K+1.
// B-frag: rows striped across lanes within a VGPR, same K split.
// C/D: vgpr i -> row (i + (lane>=16 ? 8:0)), col = lane%16.
// ---------------------------------------------------------------------------
template <int NC>
__global__ __launch_bounds__(128) void gemm_wmma_f32(
    const float* __restrict__ A, const float* __restrict__ B,
    float* __restrict__ C, int M) {
  __shared__ float lds_a[16 * 132];
  const int row0 = blockIdx.y * 16;
  const int t = threadIdx.x;

  // Cooperative A-tile load: 16 rows x 128 cols = 512 float4, 4 per thread.
  {
    const float4* A4 = (const float4*)A;
    #pragma unroll
    for (int i = 0; i < 4; ++i) {
      int idx = t + i * 128;        // 0..511
      int r = idx >> 5;             // 32 float4 per row
      int c4 = idx & 31;
      int gr = row0 + r;
      if (gr >= M) gr = M - 1;      // clamp (M is a multiple of 16 here)
      float4 v = A4[(size_t)gr * 32 + c4];
      float* dp = &lds_a[r * 132 + c4 * 4];
      dp[0] = v.x; dp[1] = v.y; dp[2] = v.z; dp[3] = v.w;
    }
  }
  __syncthreads();

  const int wave = t >> 5;
  const int lane = t & 31;
  const int col0 = (blockIdx.x * 4 + wave) * 16;
  if (col0 >= NC) return;
  const int lrow = lane & 15;
  const int khi = (lane >> 4) * 2;  // 0 for lanes 0-15, 2 for lanes 16-31

  v8f c = {};
  #pragma unroll
  for (int k0 = 0; k0 < 128; k0 += 4) {
    v2f a, b;
    a.x = lds_a[lrow * 132 + k0 + khi];
    a.y = lds_a[lrow * 132 + k0 + khi + 1];
    b.x = B[(size_t)(k0 + khi) * NC + col0 + lrow];
    b.y = B[(size_t)(k0 + khi + 1) * NC + col0 + lrow];
    c = __builtin_amdgcn_wmma_f32_16x16x4_f32(
        /*neg_a=*/false, a, /*neg_b=*/false, b,
        /*c_mod=*/(short)0, c, /*reuse_a=*/false, /*reuse_b=*/false);
  }

  const int srow = row0 + (lane >> 4) * 8;
  const int scol = col0 + lrow;
  #pragma unroll
  for (int i = 0; i < 8; ++i)
    if (srow + i < M) C[(size_t)(srow + i) * NC + scol] = c[i];
}

// ---------------------------------------------------------------------------
// Per-node attention coefficients: as[n,h] = <h[n,h,:], a_src[h,:]>, same ad.
// ---------------------------------------------------------------------------
template <int H, int CH>
__global__ void attn_coef(const float* __restrict__ hf,
                          const float* __restrict__ a_src,
                          const float* __restrict__ a_dst,
                          float* __restrict__ as, float* __restrict__ ad,
                          int N) {
  int i = blockIdx.x * blockDim.x + threadIdx.x;
  if (i >= N * H) return;
  int n = i / H, h = i - n * H;
  const float* hp = hf + (size_t)n * H * CH + h * CH;
  float s = 0.f, d = 0.f;
  #pragma unroll 8
  for (int cIdx = 0; cIdx < CH; ++cIdx) {
    float v = hp[cIdx];
    s += v * a_src[h * CH + cIdx];
    d += v * a_dst[h * CH + cIdx];
  }
  as[i] = s;
  ad[i] = d;
}

// Edge pass 1: segment max of leaky_relu(as[src]+ad[dst]) over dst.
template <int H>
__global__ void edge_max_k(const int* __restrict__ ei,
                           const float* __restrict__ as,
                           const float* __restrict__ ad,
                           unsigned* __restrict__ m, int E, int N) {
  int e = blockIdx.x * blockDim.x + threadIdx.x;
  if (e >= E + N) return;
  int s = (e < E) ? ei[e] : (e - E);
  int d = (e < E) ? ei[E + e] : (e - E);
  #pragma unroll
  for (int h = 0; h < H; ++h) {
    float x = as[s * H + h] + ad[d * H + h];
    x = (x > 0.f) ? x : NEG_SLOPE * x;
    atomicMax(&m[d * H + h], f2ord(x));
  }
}

// Edge pass 2: denom[dst] += exp(e - max[dst]).
template <int H>
__global__ void edge_sum_k(const int* __restrict__ ei,
                           const float* __restrict__ as,
                           const float* __restrict__ ad,
                           const unsigned* __restrict__ m,
                           float* __restrict__ den, int E, int N) {
  int e = blockIdx.x * blockDim.x + threadIdx.x;
  if (e >= E + N) return;
  int s = (e < E) ? ei[e] : (e - E);
  int d = (e < E) ? ei[E + e] : (e - E);
  #pragma unroll
  for (int h = 0; h < H; ++h) {
    float x = as[s * H + h] + ad[d * H + h];
    x = (x > 0.f) ? x : NEG_SLOPE * x;
    atomicAdd(&den[d * H + h], expf(x - ord2f(m[d * H + h])));
  }
}

// Edge pass 3: acc[dst, k] += alpha(e,h) * h[src, k]. One thread per (edge, k).
template <int H, int CH>
__global__ void edge_aggr_k(const int* __restrict__ ei,
                            const float* __restrict__ as,
                            const float* __restrict__ ad,
                            const unsigned* __restrict__ m,
                            const float* __restrict__ den,
                            const float* __restrict__ hf,
                            float* __restrict__ acc, int E, int N) {
  const int HC = H * CH;
  int idx = blockIdx.x * blockDim.x + threadIdx.x;
  if (idx >= (E + N) * HC) return;
  int e = idx / HC;
  int k = idx - e * HC;
  int h = k / CH;
  int s = (e < E) ? ei[e] : (e - E);
  int d = (e < E) ? ei[E + e] : (e - E);
  float x = as[s * H + h] + ad[d * H + h];
  x = (x > 0.f) ? x : NEG_SLOPE * x;
  float alpha = expf(x - ord2f(m[d * H + h])) / den[d * H + h];
  atomicAdd(&acc[(size_t)d * HC + k], alpha * hf[(size_t)s * HC + k]);
}

// out = elu(acc + bias)  (layer-1 epilogue, in place)
__global__ void bias_elu_k(float* __restrict__ acc, const float* __restrict__ b,
                           int total, int K) {
  int i = blockIdx.x * blockDim.x + threadIdx.x;
  if (i >= total) return;
  float v = acc[i] + b[i % K];
  acc[i] = (v > 0.f) ? v : (expf(v) - 1.f);
}

// out += bias  (layer-2 epilogue)
__global__ void bias_add_k(float* __restrict__ out, const float* __restrict__ b,
                           int total, int K) {
  int i = blockIdx.x * blockDim.x + threadIdx.x;
  if (i >= total) return;
  out[i] += b[i % K];
}

// ---------------------------------------------------------------------------
extern "C" void kernel_launch(void* const* d_in, const int* in_sizes, int n_in,
                              void* d_out, int out_size, void* d_ws,
                              size_t ws_size, hipStream_t stream) {
  const float* x      = (const float*)d_in[0];  // [N,128]
  const int*   ei     = (const int*)d_in[1];    // [2,E]
  const float* W1     = (const float*)d_in[2];  // [128,128]
  const float* a_src1 = (const float*)d_in[3];  // [4,32]
  const float* a_dst1 = (const float*)d_in[4];
  const float* b1     = (const float*)d_in[5];  // [128]
  const float* W2     = (const float*)d_in[6];  // [128,64]
  const float* a_src2 = (const float*)d_in[7];  // [1,64]
  const float* a_dst2 = (const float*)d_in[8];
  const float* b2     = (const float*)d_in[9];  // [64]
  float* out = (float*)d_out;                   // [N,64]

  const int N = in_sizes[0] / 128;
  const int E = in_sizes[1] / 2;
  const int Et = E + N;  // edges + self loops

  // ---- workspace layout (floats) ----
  float* ws  = (float*)d_ws;
  float* h1  = ws;                          // N*128  projected layer-1 feats
  float* as1 = h1 + (size_t)N * 128;        // N*4
  float* ad1 = as1 + (size_t)N * 4;         // N*4
  float* h2  = ad1 + (size_t)N * 4;         // N*64   projected layer-2 feats
  float* as2 = h2 + (size_t)N * 64;         // N
  float* ad2 = as2 + (size_t)N;             // N
  float* z0  = ad2 + (size_t)N;             // ---- zero-init region ----
  unsigned* m1   = (unsigned*)z0;           // N*4  (ordered-uint max)
  float*    den1 = z0 + (size_t)N * 4;      // N*4
  float*    acc1 = den1 + (size_t)N * 4;    // N*128
  unsigned* m2   = (unsigned*)(acc1 + (size_t)N * 128);  // N
  float*    den2 = (float*)m2 + (size_t)N;  // N
  const size_t zcount = (size_t)N * (4 + 4 + 128 + 1 + 1);

  hipMemsetAsync(z0, 0, zcount * sizeof(float), stream);
  hipMemsetAsync(out, 0, (size_t)out_size * sizeof(float), stream);

  const int TB = 256;
  const int rowTiles = (N + 15) / 16;

  // ============ Layer 1 ============
  gemm_wmma_f32<128><<<dim3(2, rowTiles), 128, 0, stream>>>(x, W1, h1, N);
  attn_coef<4, 32><<<(N * 4 + TB - 1) / TB, TB, 0, stream>>>(
      h1, a_src1, a_dst1, as1, ad1, N);
  edge_max_k<4><<<(Et + TB - 1) / TB, TB, 0, stream>>>(ei, as1, ad1, m1, E, N);
  edge_sum_k<4><<<(Et + TB - 1) / TB, TB, 0, stream>>>(ei, as1, ad1, m1, den1,
                                                       E, N);
  {
    long long total = (long long)Et * 128;
    edge_aggr_k<4, 32><<<(unsigned)((total + TB - 1) / TB), TB, 0, stream>>>(
        ei, as1, ad1, m1, den1, h1, acc1, E, N);
  }
  bias_elu_k<<<(N * 128 + TB - 1) / TB, TB, 0, stream>>>(acc1, b1, N * 128,
                                                         128);

  // ============ Layer 2 ============
  gemm_wmma_f32<64><<<dim3(1, rowTiles), 128, 0, stream>>>(acc1, W2, h2, N);
  attn_coef<1, 64><<<(N + TB - 1) / TB, TB, 0, stream>>>(h2, a_src2, a_dst2,
                                                         as2, ad2, N);
  edge_max_k<1><<<(Et + TB - 1) / TB, TB, 0, stream>>>(ei, as2, ad2, m2, E, N);
  edge_sum_k<1><<<(Et + TB - 1) / TB, TB, 0, stream>>>(ei, as2, ad2, m2, den2,
                                                       E, N);
  {
    long long total = (long long)Et * 64;
    edge_aggr_k<1, 64><<<(unsigned)((total + TB - 1) / TB), TB, 0, stream>>>(
        ei, as2, ad2, m2, den2, h2, out, E, N);
  }
  bias_add_k<<<(N * 64 + TB - 1) / TB, TB, 0, stream>>>(out, b2, N * 64, 64);
}